// sFastGuidedFilter_35931696398898
// MI455X (gfx1250) — compile-verified
//
#include <hip/hip_runtime.h>

// ---------------------------------------------------------------------------
// Fused fast-guided-filter + 8-directional residual selection, single pass.
//   lr_x, lr_y : [1,3,2048,2048] f32   (hr_x is unused by the reference)
//   out        : [1,3,2048,2048] f32 = clip(trunc(best_residual + lr_x),0,255)
// Tile: 32x32 output per block; 36x36 input halo (box needs +-2 total reach).
// Halo tiles are DMA'd into LDS with GLOBAL_LOAD_ASYNC_TO_LDS_B32 (per-lane
// clamped global addresses implement the replicate padding directly).
// Output uses non-temporal stores (write-once data, keep L2 for halo reuse).
// ---------------------------------------------------------------------------

#define TILE  32
#define EXT   34            // TILE + 2  (A,b region: output +- 1)
#define IN    36            // TILE + 4  (input region: output +- 2)
#define IMG_H 2048
#define IMG_W 2048
#define GF_EPS 1e-4f

#ifndef __has_builtin
#define __has_builtin(x) 0
#endif

#if __has_builtin(__builtin_amdgcn_global_load_async_to_lds_b32)
#define HAVE_ASYNC_LDS 1
#else
#define HAVE_ASYNC_LDS 0
#warning "CDNA5 probe: __builtin_amdgcn_global_load_async_to_lds_b32 MISSING (sync fallback)"
#endif

#if __has_builtin(__builtin_amdgcn_s_wait_asynccnt)
#define HAVE_WAIT_ASYNC 1
#else
#define HAVE_WAIT_ASYNC 0
#endif

// Builtin signature (from clang diagnostic): (int AS1*, int AS3*, imm, imm)
typedef __attribute__((address_space(1))) int gint_t;
typedef __attribute__((address_space(3))) int lint_t;

__device__ __forceinline__ void copy_g2lds_f32(const float* __restrict__ g, float* l) {
#if HAVE_ASYNC_LDS
  __builtin_amdgcn_global_load_async_to_lds_b32(
      (gint_t*)(void*)g, (lint_t*)l, /*offset=*/0, /*cpol=*/0);
#else
  *l = __ldg(g);
#endif
}

__device__ __forceinline__ void wait_async_lds() {
#if HAVE_ASYNC_LDS
#if HAVE_WAIT_ASYNC
  __builtin_amdgcn_s_wait_asynccnt(0);
#else
  asm volatile("s_wait_asynccnt 0" ::: "memory");
#endif
#endif
}

__device__ __forceinline__ int clampi(int v, int lo, int hi) {
  return v < lo ? lo : (v > hi ? hi : v);
}

__global__ __launch_bounds__(256)
void guided_dir_kernel(const float* __restrict__ lrx,
                       const float* __restrict__ lry,
                       float* __restrict__ out)
{
  __shared__ float sX[IN * IN];    // guide tile  (edge-clamped coords)
  __shared__ float sY[IN * IN];    // target tile (edge-clamped coords)
  __shared__ float sA[EXT * EXT];  // guided-filter A (0 outside image)
  __shared__ float sB[EXT * EXT];  // guided-filter b (0 outside image)

  const int c   = blockIdx.z;
  const int ox  = blockIdx.x * TILE;
  const int oy  = blockIdx.y * TILE;
  const int tid = threadIdx.x;

  const float* X = lrx + (size_t)c * IMG_H * IMG_W;
  const float* Y = lry + (size_t)c * IMG_H * IMG_W;

  // ---- Stage 1: async DMA the 36x36 halo tiles into LDS ------------------
  // Replicate padding == per-lane coordinate clamp; async-to-LDS supports
  // fully divergent per-lane (global addr, LDS addr) pairs.
  // Incremental (ix,iy) update: step 256 == 7*IN + 4.
  {
    int ix = tid % IN;
    int iy = tid / IN;
    for (int i = tid; i < IN * IN; i += 256) {
      const int gx = clampi(ox - 2 + ix, 0, IMG_W - 1);
      const int gy = clampi(oy - 2 + iy, 0, IMG_H - 1);
      const size_t go = (size_t)gy * IMG_W + gx;
      copy_g2lds_f32(X + go, &sX[i]);
      copy_g2lds_f32(Y + go, &sY[i]);
      ix += 4; iy += 7;
      if (ix >= IN) { ix -= IN; iy += 1; }
    }
  }
  wait_async_lds();
  __syncthreads();

  // ---- Stage 2: guided-filter A,b on the 34x34 extended region -----------
  const float inv9 = 1.0f / 9.0f;
  {
    int ex = tid % EXT;
    int ey = tid / EXT;
    // step 256 == 7*EXT + 18
    for (int i = tid; i < EXT * EXT; i += 256) {
      float sx = 0.f, sy = 0.f, sxy = 0.f, sxx = 0.f;
#pragma unroll
      for (int dy = 0; dy < 3; ++dy) {
#pragma unroll
        for (int dx = 0; dx < 3; ++dx) {
          const float xv = sX[(ey + dy) * IN + (ex + dx)];
          const float yv = sY[(ey + dy) * IN + (ex + dx)];
          sx += xv;
          sy += yv;
          sxy = fmaf(xv, yv, sxy);
          sxx = fmaf(xv, xv, sxx);
        }
      }
      const float mx  = sx  * inv9;
      const float my  = sy  * inv9;
      const float mxy = sxy * inv9;
      const float mxx = sxx * inv9;
      const float A = (mxy - mx * my) / (mxx - mx * mx + GF_EPS);
      const float b = my - A * mx;
      // Directional convs use ZERO padding at the image boundary.
      const int gx = ox - 1 + ex;
      const int gy = oy - 1 + ey;
      const bool inimg = (gx >= 0) && (gx < IMG_W) && (gy >= 0) && (gy < IMG_H);
      sA[i] = inimg ? A : 0.f;
      sB[i] = inimg ? b : 0.f;
      ex += 18; ey += 7;
      if (ex >= EXT) { ex -= EXT; ey += 1; }
    }
  }
  __syncthreads();

  // ---- Stage 3: 8 directional residuals + argmin, 1x4 strip per thread ---
  const int tx  = tid & 31;   // output column within tile (one wave = one row group)
  const int tg  = tid >> 5;   // 0..7
  const int ty0 = tg * 4;     // first output row of this thread's strip

  // Horizontal partial sums for 6 consecutive ext rows (rows ty0..ty0+5).
  float hAL[6], hAR[6], hA3[6], hBL[6], hBR[6], hB3[6];
#pragma unroll
  for (int j = 0; j < 6; ++j) {
    const int er = ty0 + j;                     // ext row
    const float* ra = &sA[er * EXT + tx];       // ext cols tx..tx+2
    const float a0 = ra[0], a1 = ra[1], a2 = ra[2];
    hAL[j] = a0 + a1;
    hAR[j] = a1 + a2;
    hA3[j] = hAL[j] + a2;
    const float* rb = &sB[er * EXT + tx];
    const float b0 = rb[0], b1 = rb[1], b2 = rb[2];
    hBL[j] = b0 + b1;
    hBR[j] = b1 + b2;
    hB3[j] = hBL[j] + b2;
  }

  const float n6 = 1.0f / 6.0f;
  const float n4 = 0.25f;

#pragma unroll
  for (int rr = 0; rr < 4; ++rr) {
    const int j0 = rr, j1 = rr + 1, j2 = rr + 2;
    const float im = sX[(ty0 + rr + 2) * IN + (tx + 2)];

    // Directional window sums from shared separable partials.
    const float nwA = hAL[j0] + hAL[j1], swA = hAL[j1] + hAL[j2];
    const float neA = hAR[j0] + hAR[j1], seA = hAR[j1] + hAR[j2];
    const float lA  = nwA + hAL[j2],     rA  = neA + hAR[j2];
    const float uA  = hA3[j0] + hA3[j1], dA  = hA3[j1] + hA3[j2];

    const float nwB = hBL[j0] + hBL[j1], swB = hBL[j1] + hBL[j2];
    const float neB = hBR[j0] + hBR[j1], seB = hBR[j1] + hBR[j2];
    const float lB  = nwB + hBL[j2],     rB  = neB + hBR[j2];
    const float uB  = hB3[j0] + hB3[j1], dB  = hB3[j1] + hB3[j2];

    // Reference direction order: L, R, U, D, NW, NE, SW, SE
    const float sa[8] = { lA, rA, uA, dA, nwA, neA, swA, seA };
    const float sb[8] = { lB, rB, uB, dB, nwB, neB, swB, seB };
    const float nm[8] = { n6, n6, n6, n6, n4, n4, n4, n4 };

    float bestd = 0.f, besta = 0.f;
#pragma unroll
    for (int k = 0; k < 8; ++k) {
      // d = nm*(sumA*im + sumB) - im  ==  convA*im + convb - im
      const float d  = fmaf(nm[k], fmaf(sa[k], im, sb[k]), -im);
      const float ad = fabsf(d);
      if (k == 0 || ad < besta) { besta = ad; bestd = d; }  // first-min, like argmin
    }

    float res = truncf(bestd + im);
    res = fminf(fmaxf(res, 0.f), 255.f);
    // Write-once output: non-temporal store keeps L2 free for halo reuse.
    __builtin_nontemporal_store(
        res, &out[((size_t)c * IMG_H + (oy + ty0 + rr)) * IMG_W + (ox + tx)]);
  }
}

extern "C" void kernel_launch(void* const* d_in, const int* in_sizes, int n_in,
                              void* d_out, int out_size, void* d_ws, size_t ws_size,
                              hipStream_t stream) {
  const float* lrx = (const float*)d_in[0];
  const float* lry = (const float*)d_in[1];
  // d_in[2] (hr_x) is unused by the reference computation.
  (void)in_sizes; (void)n_in; (void)out_size; (void)d_ws; (void)ws_size;
  float* out = (float*)d_out;

  dim3 grid(IMG_W / TILE, IMG_H / TILE, 3);
  guided_dir_kernel<<<grid, dim3(256), 0, stream>>>(lrx, lry, out);
}